// GINNet_50465865728555
// MI455X (gfx1250) — compile-verified
//
#include <hip/hip_runtime.h>
#include <hip/hip_bf16.h>

typedef float v2f __attribute__((ext_vector_type(2)));
typedef float v8f __attribute__((ext_vector_type(8)));
typedef int   v4i __attribute__((ext_vector_type(4)));

#define BN_EPS 1e-5f

#if __has_builtin(__builtin_amdgcn_global_load_async_to_lds_b128)
#define HAVE_ASYNC_LDS 1
#else
#define HAVE_ASYNC_LDS 0
#endif

__device__ __forceinline__ void wait_async_zero() {
#if __has_builtin(__builtin_amdgcn_s_wait_asynccnt)
    __builtin_amdgcn_s_wait_asynccnt(0);
#else
    asm volatile("s_wait_asynccnt 0x0" ::: "memory");
#endif
}

// ---------------------------------------------------------------- zero
__global__ void k_zero(float* p, long n) {
    long i = (long)blockIdx.x * blockDim.x + threadIdx.x;
    long stride = (long)gridDim.x * blockDim.x;
    for (; i < n; i += stride) p[i] = 0.0f;
}

// ---------------------------------------------------------------- embed: h = feat @ emb_w + emb_b
__global__ void k_embed(const float* __restrict__ feat, const float* __restrict__ w,
                        const float* __restrict__ b, float* __restrict__ h,
                        int N, int IN, int H) {
    long total = (long)N * H;
    long i = (long)blockIdx.x * blockDim.x + threadIdx.x;
    long stride = (long)gridDim.x * blockDim.x;
    for (; i < total; i += stride) {
        int r = (int)(i / H);
        int c = (int)(i % H);
        float acc = b[c];
        const float* fr = feat + (long)r * IN;
        for (int k = 0; k < IN; ++k) acc += fr[k] * w[(long)k * H + c];
        h[i] = acc;
    }
}

// ---------------------------------------------------------------- per-graph node counts
__global__ void k_count(const int* __restrict__ gid, float* __restrict__ cnt, int N) {
    long i = (long)blockIdx.x * blockDim.x + threadIdx.x;
    long stride = (long)gridDim.x * blockDim.x;
    for (; i < N; i += stride) atomicAdd(&cnt[gid[i]], 1.0f);
}

// ---------------------------------------------------------------- pool accumulate (no residual)
__global__ void k_pool(const float* __restrict__ h, const int* __restrict__ gid,
                       float* __restrict__ pool, int N, int H) {
    long total = (long)N * H;
    long i = (long)blockIdx.x * blockDim.x + threadIdx.x;
    long stride = (long)gridDim.x * blockDim.x;
    for (; i < total; i += stride) {
        int r = (int)(i / H);
        int c = (int)(i % H);
        atomicAdd(&pool[(long)gid[r] * H + c], h[i]);
    }
}

// ---------------------------------------------------------------- readout: score (+)= (pool/cnt) @ pw + pb
__global__ void k_score(const float* __restrict__ pool, const float* __restrict__ cnt,
                        const float* __restrict__ pw, const float* __restrict__ pb,
                        float* __restrict__ score, int B, int H, int C, int beta) {
    int i = blockIdx.x * blockDim.x + threadIdx.x;
    if (i >= B * C) return;
    int b = i / C;
    int c = i % C;
    float inv = 1.0f / fmaxf(cnt[b], 1.0f);
    float acc = pb[c];
    const float* pr = pool + (long)b * H;
    for (int k = 0; k < H; ++k) acc += pr[k] * inv * pw[(long)k * C + c];
    score[i] = beta ? (score[i] + acc) : acc;
}

// ---------------------------------------------------------------- edge scatter: neigh[dst] += h[src]
__global__ void k_edge(const float* __restrict__ h, const int* __restrict__ src,
                       const int* __restrict__ dst, float* __restrict__ neigh,
                       long E, int H) {
    int vecs = H >> 2;  // float4 chunks per row
    long total = E * vecs;
    long i = (long)blockIdx.x * blockDim.x + threadIdx.x;
    long stride = (long)gridDim.x * blockDim.x;
    for (; i < total; i += stride) {
        long e = i / vecs;
        int q = (int)(i % vecs) << 2;
        const float4 v = *(const float4*)(h + (long)src[e] * H + q);
        float* o = neigh + (long)dst[e] * H + q;
        atomicAdd(o + 0, v.x);
        atomicAdd(o + 1, v.y);
        atomicAdd(o + 2, v.z);
        atomicAdd(o + 3, v.w);
    }
}

// ---------------------------------------------------------------- BN column stats (sum, sumsq)
#define BN_ROWS 8192
__global__ void k_bnstats(const float* __restrict__ x, int N, int H, float* __restrict__ sums) {
    int c = blockIdx.x;                 // column
    int r0 = blockIdx.y * BN_ROWS;
    int rend = r0 + BN_ROWS; if (rend > N) rend = N;
    float s = 0.0f, s2 = 0.0f;
    for (int r = r0 + threadIdx.x; r < rend; r += 256) {
        float v = x[(long)r * H + c];
        s += v; s2 += v * v;
    }
    __shared__ float ls[256], ls2[256];
    ls[threadIdx.x] = s; ls2[threadIdx.x] = s2;
    __syncthreads();
    for (int o = 128; o > 0; o >>= 1) {
        if (threadIdx.x < o) { ls[threadIdx.x] += ls[threadIdx.x + o]; ls2[threadIdx.x] += ls2[threadIdx.x + o]; }
        __syncthreads();
    }
    if (threadIdx.x == 0) {
        atomicAdd(&sums[c], ls[0]);
        atomicAdd(&sums[H + c], ls2[0]);
    }
}

// ---------------------------------------------------------------- BN finalize -> scale/shift
__global__ void k_bnfin(const float* __restrict__ sums, const float* __restrict__ g,
                        const float* __restrict__ be, float* __restrict__ scale,
                        float* __restrict__ shift, int N, int H) {
    int c = blockIdx.x * blockDim.x + threadIdx.x;
    if (c >= H) return;
    float invN = 1.0f / (float)N;
    float mean = sums[c] * invN;
    float var  = sums[H + c] * invN - mean * mean;
    float sc = g[c] * rsqrtf(var + BN_EPS);
    scale[c] = sc;
    shift[c] = be[c] - mean * sc;
}

// ---------------------------------------------------------------- y = relu(scA*z + shA), fused stats of y
__global__ void k_ystats(const float* __restrict__ z, const float* __restrict__ scA,
                         const float* __restrict__ shA, float* __restrict__ y,
                         float* __restrict__ sums, int N, int H) {
    int c = blockIdx.x;
    float sa = scA[c], sha = shA[c];
    int r0 = blockIdx.y * BN_ROWS;
    int rend = r0 + BN_ROWS; if (rend > N) rend = N;
    float s = 0.0f, s2 = 0.0f;
    for (int r = r0 + threadIdx.x; r < rend; r += 256) {
        long idx = (long)r * H + c;
        float v = fmaxf(z[idx] * sa + sha, 0.0f);
        y[idx] = v;
        s += v; s2 += v * v;
    }
    __shared__ float ls[256], ls2[256];
    ls[threadIdx.x] = s; ls2[threadIdx.x] = s2;
    __syncthreads();
    for (int o = 128; o > 0; o >>= 1) {
        if (threadIdx.x < o) { ls[threadIdx.x] += ls[threadIdx.x + o]; ls2[threadIdx.x] += ls2[threadIdx.x + o]; }
        __syncthreads();
    }
    if (threadIdx.x == 0) {
        atomicAdd(&sums[c], ls[0]);
        atomicAdd(&sums[H + c], ls2[0]);
    }
}

// ---------------------------------------------------------------- h += relu(scB*y + shB); pool += h
__global__ void k_residual_pool(float* __restrict__ h, const float* __restrict__ y,
                                const float* __restrict__ scB, const float* __restrict__ shB,
                                const int* __restrict__ gid, float* __restrict__ pool,
                                int N, int H) {
    long total = (long)N * H;
    long i = (long)blockIdx.x * blockDim.x + threadIdx.x;
    long stride = (long)gridDim.x * blockDim.x;
    for (; i < total; i += stride) {
        int r = (int)(i / H);
        int c = (int)(i % H);
        float v = h[i] + fmaxf(y[i] * scB[c] + shB[c], 0.0f);
        h[i] = v;
        atomicAdd(&pool[(long)gid[r] * H + c], v);
    }
}

// ================================================================ WMMA GEMMs
// MODE 1: A = (1+eps[layer])*in0 + in1        (GIN eps-combine, in1 = neigh)
// MODE 2: A = relu(scale[k]*in0 + shift[k])   (fused BN + relu on GEMM input)
// Operand layouts per CDNA5 ISA 7.12.2 (32-bit A 16x4 / B 4x16 / C/D 16x16).

// ---- specialized H==80 path: LDS-staged, fully unrolled K (20 WMMAs/tile) ----
template <int MODE>
__global__ void __launch_bounds__(256)
k_gemm80(const float* __restrict__ in0, const float* __restrict__ in1,
         const float* __restrict__ epsArr, int layer,
         const float* __restrict__ scale, const float* __restrict__ shift,
         const float* __restrict__ W, const float* __restrict__ bias,
         float* __restrict__ out, int N) {
    constexpr int H    = 80;
    constexpr int AST  = 84;   // padded A stride: bank step 20, conflict-free over 16 lanes
    constexpr int WST  = 82;   // padded Wt stride: bank step 18, conflict-free over 16/32 lanes
    constexpr int ROWS = 128;  // 8 waves x 16 rows

    __shared__ __align__(16) float Al[ROWS * AST];   // A slab (prologue applied)
    __shared__ __align__(16) float Wt[H * WST];      // W transposed: Wt[n][k] = W[k][n]
#if HAVE_ASYNC_LDS
    __shared__ __align__(16) float Wraw[H * H];      // async-copied raw W
#endif

    const int tid     = threadIdx.x;
    const int rowBase = blockIdx.x * ROWS;

    float alpha = 1.0f;
    if (MODE == 1) alpha = 1.0f + epsArr[layer];

    // hint: prefetch next block's A slab (gfx1250 global_prefetch path)
    if (tid == 0) __builtin_prefetch(in0 + (long)(rowBase + ROWS) * H, 0, 0);

#if HAVE_ASYNC_LDS
    // ---- async global->LDS raw copy of W (ASYNCcnt path); overlaps with A staging
    for (int it = tid; it < (H * H) / 4; it += 256) {
        __builtin_amdgcn_global_load_async_to_lds_b128(
            (v4i*)(W + it * 4),
            (v4i*)&Wraw[it * 4],
            0, 0);
    }
#else
    // ---- synchronous staging of W transposed into LDS (coalesced global reads)
    for (int it = tid; it < H * H; it += 256) {
        int k = it / H;
        int n = it - k * H;
        Wt[n * WST + k] = W[it];
    }
#endif

    // ---- stage A slab with fused elementwise prologue (float4 coalesced)
    for (int it = tid; it < ROWS * (H / 4); it += 256) {
        int rl = it / (H / 4);
        int kq = (it - rl * (H / 4)) * 4;
        int r  = rowBase + rl;
        float4 v = {0.0f, 0.0f, 0.0f, 0.0f};
        if (r < N) {
            v = *(const float4*)(in0 + (long)r * H + kq);
            if (MODE == 1) {
                const float4 nb = *(const float4*)(in1 + (long)r * H + kq);
                v.x = alpha * v.x + nb.x;
                v.y = alpha * v.y + nb.y;
                v.z = alpha * v.z + nb.z;
                v.w = alpha * v.w + nb.w;
            } else if (MODE == 2) {
                const float4 sc = *(const float4*)(scale + kq);
                const float4 sh = *(const float4*)(shift + kq);
                v.x = fmaxf(v.x * sc.x + sh.x, 0.0f);
                v.y = fmaxf(v.y * sc.y + sh.y, 0.0f);
                v.z = fmaxf(v.z * sc.z + sh.z, 0.0f);
                v.w = fmaxf(v.w * sc.w + sh.w, 0.0f);
            }
        }
        *(float4*)&Al[rl * AST + kq] = v;  // 16B-aligned ds_store_b128
    }

#if HAVE_ASYNC_LDS
    wait_async_zero();      // this wave's async LDS writes complete
    __syncthreads();        // all waves' writes visible
    // in-LDS transpose Wraw -> padded Wt (read stride 1, write bank-step 18: conflict-free)
    for (int it = tid; it < H * H; it += 256) {
        int k = it / H;
        int n = it - k * H;
        Wt[n * WST + k] = Wraw[it];
    }
#endif
    __syncthreads();

    const int lane = tid & 31;
    const int half = lane >> 4;   // 0: lanes 0-15, 1: lanes 16-31
    const int m    = lane & 15;
    const int wave = tid >> 5;
    const int rl0  = wave * 16;
    const float* __restrict__ Arow = &Al[(rl0 + m) * AST];

    for (int ct = 0; ct < H / 16; ++ct) {
        const int col = ct * 16 + m;
        const float* __restrict__ Bcol = &Wt[col * WST];
        v8f acc = {};
        #pragma unroll
        for (int kt = 0; kt < H / 4; ++kt) {
            const int ka = kt * 4 + 2 * half;     // A VGPR0 -> K=ka, VGPR1 -> K=ka+1
            v2f a; a.x = Arow[ka]; a.y = Arow[ka + 1];   // ds_load_b64 (pairs -> ds_load_2addr_b64)
            v2f b; b.x = Bcol[ka]; b.y = Bcol[ka + 1];   // ds_load_b64 (Wt transposed)
            acc = __builtin_amdgcn_wmma_f32_16x16x4_f32(
                false, a, false, b, (short)0, acc, false, false);
        }
        const float bv = bias[col];
        #pragma unroll
        for (int j = 0; j < 8; ++j) {
            int r = rowBase + rl0 + j + 8 * half;  // C/D: VGPR j -> M=j / M=j+8
            if (r < N) out[(long)r * H + col] = acc[j] + bv;
        }
    }
}

// ---- generic fallback (H % 16 == 0), direct global loads, one wave per tile ----
template <int MODE>
__global__ void k_gemm_wmma(const float* __restrict__ in0, const float* __restrict__ in1,
                            const float* __restrict__ epsArr, int layer,
                            const float* __restrict__ scale, const float* __restrict__ shift,
                            const float* __restrict__ W, const float* __restrict__ bias,
                            float* __restrict__ out, int N, int H) {
    const int lane = threadIdx.x & 31;
    const int half = lane >> 4;
    const int m    = lane & 15;
    const int row0 = blockIdx.x * 16;
    const int col0 = blockIdx.y * 16;
    const int col  = col0 + m;

    float alpha = 1.0f;
    if (MODE == 1) alpha = 1.0f + epsArr[layer];

    v8f acc = {};
    const int ra = row0 + m;
    for (int k0 = 0; k0 < H; k0 += 4) {
        const int ka = k0 + 2 * half;
        float a0 = 0.0f, a1 = 0.0f;
        if (ra < N) {
            long base = (long)ra * H + ka;
            a0 = in0[base];
            a1 = in0[base + 1];
            if (MODE == 1) {
                a0 = alpha * a0 + in1[base];
                a1 = alpha * a1 + in1[base + 1];
            } else if (MODE == 2) {
                a0 = fmaxf(a0 * scale[ka]     + shift[ka],     0.0f);
                a1 = fmaxf(a1 * scale[ka + 1] + shift[ka + 1], 0.0f);
            }
        }
        v2f a; a.x = a0; a.y = a1;
        v2f b; b.x = W[(long)ka * H + col]; b.y = W[(long)(ka + 1) * H + col];
        acc = __builtin_amdgcn_wmma_f32_16x16x4_f32(
            false, a, false, b, (short)0, acc, false, false);
    }

    const float bv = bias[col];
    #pragma unroll
    for (int j = 0; j < 8; ++j) {
        int r = row0 + j + 8 * half;
        if (r < N) out[(long)r * H + col] = acc[j] + bv;
    }
}

// ---------------------------------------------------------------- host side
static inline int gblk(long total, int bs) {
    long b = (total + bs - 1) / bs;
    if (b > 1048576) b = 1048576;
    return (int)b;
}

template <int MODE>
static void launch_gemm(const float* in0, const float* in1,
                        const float* epsArr, int layer,
                        const float* scale, const float* shift,
                        const float* W, const float* bias,
                        float* out, int N, int H, hipStream_t stream) {
    if (H == 80) {
        k_gemm80<MODE><<<(N + 127) / 128, 256, 0, stream>>>(
            in0, in1, epsArr, layer, scale, shift, W, bias, out, N);
    } else {
        dim3 g((N + 15) / 16, H / 16);
        k_gemm_wmma<MODE><<<g, 32, 0, stream>>>(
            in0, in1, epsArr, layer, scale, shift, W, bias, out, N, H);
    }
}

extern "C" void kernel_launch(void* const* d_in, const int* in_sizes, int n_in,
                              void* d_out, int out_size, void* d_ws, size_t ws_size,
                              hipStream_t stream) {
    // Inputs (setup_inputs order)
    const float* feat   = (const float*)d_in[0];
    const int*   src    = (const int*)d_in[1];
    const int*   dst    = (const int*)d_in[2];
    const int*   gid    = (const int*)d_in[3];
    const float* emb_w  = (const float*)d_in[4];
    const float* emb_b  = (const float*)d_in[5];
    const float* w1     = (const float*)d_in[6];
    const float* b1     = (const float*)d_in[7];
    const float* g1     = (const float*)d_in[8];
    const float* be1    = (const float*)d_in[9];
    const float* w2     = (const float*)d_in[10];
    const float* b2     = (const float*)d_in[11];
    const float* g2a    = (const float*)d_in[12];
    const float* be2a   = (const float*)d_in[13];
    const float* g2b    = (const float*)d_in[14];
    const float* be2b   = (const float*)d_in[15];
    const float* eps    = (const float*)d_in[16];
    const float* pred_w = (const float*)d_in[17];
    const float* pred_b = (const float*)d_in[18];

    // Dimensions (derived on host; no device reads)
    const int  L  = in_sizes[16];
    const int  H  = in_sizes[5];
    const int  C  = in_sizes[18] / (L + 1);
    const int  IN = in_sizes[4] / H;
    const int  N  = in_sizes[3];
    const long E  = in_sizes[1];
    const int  B  = out_size / C;
    const long NH = (long)N * H;

    float* score = (float*)d_out;

    // Workspace carve (floats)
    float* ws    = (float*)d_ws;
    float* h     = ws;            ws += NH;
    float* t1    = ws;            ws += NH;   // GEMM1 out / y buffer
    float* z2    = ws;            ws += NH;   // GEMM2 out
    float* neigh = ws;            ws += NH;
    float* pool  = ws;            ws += (long)B * H;
    float* cnt   = ws;            ws += B;
    float* sums  = ws;            ws += 2 * H;  // [sum | sumsq]
    float* scA   = ws;            ws += H;
    float* shA   = ws;            ws += H;
    float* scB   = ws;            ws += H;
    float* shB   = ws;            ws += H;
    (void)ws_size;

    const dim3 bnGrid(H, (N + BN_ROWS - 1) / BN_ROWS);
    const int  bnFinBlk = 64;
    const int  bnFinGrd = (H + bnFinBlk - 1) / bnFinBlk;

    // ---- embedding
    k_embed<<<gblk(NH, 256), 256, 0, stream>>>(feat, emb_w, emb_b, h, N, IN, H);

    // ---- per-graph counts (constant across layers)
    k_zero<<<gblk(B, 256), 256, 0, stream>>>(cnt, B);
    k_count<<<gblk(N, 256), 256, 0, stream>>>(gid, cnt, N);

    // ---- layer-0 readout
    k_zero<<<gblk((long)B * H, 256), 256, 0, stream>>>(pool, (long)B * H);
    k_pool<<<gblk(NH, 256), 256, 0, stream>>>(h, gid, pool, N, H);
    k_score<<<(B * C + 255) / 256, 256, 0, stream>>>(pool, cnt, pred_w, pred_b, score, B, H, C, /*beta=*/0);

    for (int i = 0; i < L; ++i) {
        const float* W1 = w1 + (long)i * H * H;
        const float* B1 = b1 + (long)i * H;
        const float* W2 = w2 + (long)i * H * H;
        const float* B2 = b2 + (long)i * H;

        // neighbor sum aggregation
        k_zero<<<gblk(NH, 256), 256, 0, stream>>>(neigh, NH);
        k_edge<<<gblk(E * (H / 4), 256), 256, 0, stream>>>(h, src, dst, neigh, E, H);

        // t1 = ((1+eps)*h + neigh) @ W1 + b1   (fused combine; WMMA)
        launch_gemm<1>(h, neigh, eps, i, nullptr, nullptr, W1, B1, t1, N, H, stream);

        // BN1 stats + finalize
        k_zero<<<1, 256, 0, stream>>>(sums, 2 * H);
        k_bnstats<<<bnGrid, 256, 0, stream>>>(t1, N, H, sums);
        k_bnfin<<<bnFinGrd, bnFinBlk, 0, stream>>>(sums, g1 + (long)i * H, be1 + (long)i * H,
                                                   scA, shA, N, H);

        // z2 = relu(BN1(t1)) @ W2 + b2   (fused BN+relu in A-load; WMMA)
        launch_gemm<2>(t1, nullptr, nullptr, 0, scA, shA, W2, B2, z2, N, H, stream);

        // BN2a stats + finalize
        k_zero<<<1, 256, 0, stream>>>(sums, 2 * H);
        k_bnstats<<<bnGrid, 256, 0, stream>>>(z2, N, H, sums);
        k_bnfin<<<bnFinGrd, bnFinBlk, 0, stream>>>(sums, g2a + (long)i * H, be2a + (long)i * H,
                                                   scA, shA, N, H);

        // y = relu(BN2a(z2)) with fused stats of y, then BN2b finalize
        k_zero<<<1, 256, 0, stream>>>(sums, 2 * H);
        k_ystats<<<bnGrid, 256, 0, stream>>>(z2, scA, shA, t1 /*y*/, sums, N, H);
        k_bnfin<<<bnFinGrd, bnFinBlk, 0, stream>>>(sums, g2b + (long)i * H, be2b + (long)i * H,
                                                   scB, shB, N, H);

        // residual + mean-pool accumulate
        k_zero<<<gblk((long)B * H, 256), 256, 0, stream>>>(pool, (long)B * H);
        k_residual_pool<<<gblk(NH, 256), 256, 0, stream>>>(h, t1, scB, shB, gid, pool, N, H);

        // readout for this layer
        k_score<<<(B * C + 255) / 256, 256, 0, stream>>>(pool, cnt,
                                                         pred_w + (long)(i + 1) * H * C,
                                                         pred_b + (long)(i + 1) * C,
                                                         score, B, H, C, /*beta=*/1);
    }
}